// GCN_65068754534587
// MI455X (gfx1250) — compile-verified
//
#include <hip/hip_runtime.h>
#include <hip/hip_bf16.h>

#define NUM_GRAPHS 512

typedef __attribute__((ext_vector_type(2))) float v2f;
typedef __attribute__((ext_vector_type(8))) float v8f;

// ---------------- degree / norm ----------------

__global__ void k_deg_init(unsigned int* __restrict__ deg, int N) {
    int i = blockIdx.x * blockDim.x + threadIdx.x;
    if (i < N) deg[i] = 1u;   // self-loop
}

__global__ void k_deg_edges(const int* __restrict__ dst, unsigned int* __restrict__ deg, int E) {
    int e = blockIdx.x * blockDim.x + threadIdx.x;
    if (e < E) atomicAdd(&deg[dst[e]], 1u);
}

__global__ void k_dinv(float* __restrict__ dinv_f, int N) {
    int i = blockIdx.x * blockDim.x + threadIdx.x;
    if (i >= N) return;
    unsigned int d = ((const unsigned int*)dinv_f)[i];   // in-place u32 -> f32
    dinv_f[i] = (d > 0u) ? rsqrtf((float)d) : 0.0f;
}

// ---------------- layer-1 GEMM: h1[N,16] = x[N,3(pad4)] @ W1[3(pad4),16] ----------------
// V_WMMA_F32_16X16X4_F32; one wave computes 16 node-rows.
// Fused: also writes the self-loop-initialized accumulator a1 = dinv^2 * h1.

__global__ void k_gemm1_wmma(const float* __restrict__ x, const float* __restrict__ W1,
                             const float* __restrict__ dinv,
                             float* __restrict__ h1, float* __restrict__ a1, int N) {
    int wave = (blockIdx.x * blockDim.x + threadIdx.x) >> 5;
    int lane = threadIdx.x & 31;
    int m0 = wave * 16;
    if (m0 >= N) return;                 // wave-uniform
    const bool full = (m0 + 16 <= N);    // wave-uniform fast path (no per-lane guards)
    int hs = lane >> 4;                  // 0: lanes 0-15 (K=0,1), 1: lanes 16-31 (K=2,3)
    int lm = lane & 15;                  // A: row index; B/D: column index
    int node = m0 + lm;

    // A: 16x4, lane = row; vgpr0 = K(2*hs), vgpr1 = K(2*hs+1); K=3 padded to 0
    float a0 = 0.f, a1v = 0.f;
    if (full || node < N) {
        const float* xr = x + (size_t)node * 3;
        if (hs == 0) { a0 = xr[0]; a1v = xr[1]; }
        else         { a0 = xr[2]; a1v = 0.f;   }
    }
    v2f A; A.x = a0; A.y = a1v;

    // B: 4x16, lane = col; vgpr0 = row 2*hs, vgpr1 = row 2*hs+1 (row 3 padded)
    float b0, b1;
    if (hs == 0) { b0 = W1[0 * 16 + lm]; b1 = W1[1 * 16 + lm]; }
    else         { b0 = W1[2 * 16 + lm]; b1 = 0.f;             }
    v2f B; B.x = b0; B.y = b1;

    v8f C = {};
    C = __builtin_amdgcn_wmma_f32_16x16x4_f32(false, A, false, B, (short)0, C, false, false);

    // D: lane col = lm, VGPR r -> row m0 + r + 8*hs
    if (full) {
        #pragma unroll
        for (int r = 0; r < 8; ++r) {
            int row = m0 + r + hs * 8;
            float v = C[r];
            float di = dinv[row];
            h1[(size_t)row * 16 + lm] = v;
            a1[(size_t)row * 16 + lm] = di * di * v;
        }
    } else {
        #pragma unroll
        for (int r = 0; r < 8; ++r) {
            int row = m0 + r + hs * 8;
            if (row < N) {
                float v = C[r];
                float di = dinv[row];
                h1[(size_t)row * 16 + lm] = v;
                a1[(size_t)row * 16 + lm] = di * di * v;
            }
        }
    }
}

// ---------------- layer-1 edge scatter ----------------

__global__ void k_edges16(const int* __restrict__ src, const int* __restrict__ dst,
                          const float* __restrict__ dinv, const float* __restrict__ h,
                          float* __restrict__ acc, int E) {
    int e = blockIdx.x * blockDim.x + threadIdx.x;
    if (e >= E) return;
    int s = src[e], d = dst[e];
    float norm = dinv[s] * dinv[d];
    const float4* hs4 = (const float4*)(h + (size_t)s * 16);
    float* out = acc + (size_t)d * 16;
    #pragma unroll
    for (int q = 0; q < 4; ++q) {
        float4 v = hs4[q];
        atomicAdd(out + q * 4 + 0, v.x * norm);
        atomicAdd(out + q * 4 + 1, v.y * norm);
        atomicAdd(out + q * 4 + 2, v.z * norm);
        atomicAdd(out + q * 4 + 3, v.w * norm);
    }
}

// ---------------- layer-2 GEMM: h2[N,2] = relu(a1+b1)[N,16] @ W2[16,2(pad16)] ----------------
// 4 chained V_WMMA_F32_16X16X4_F32 (K = 16). Bias+ReLU fused into the A load;
// also writes the self-loop-initialized accumulator a2 = dinv^2 * h2.

__global__ void k_gemm2_wmma(const float* __restrict__ a1, const float* __restrict__ b1,
                             const float* __restrict__ W2, const float* __restrict__ dinv,
                             float* __restrict__ h2, float* __restrict__ a2, int N) {
    int wave = (blockIdx.x * blockDim.x + threadIdx.x) >> 5;
    int lane = threadIdx.x & 31;
    int m0 = wave * 16;
    if (m0 >= N) return;
    const bool full = (m0 + 16 <= N);
    int hs = lane >> 4, lm = lane & 15;
    int node = m0 + lm;

    v8f C = {};
    #pragma unroll
    for (int kk = 0; kk < 4; ++kk) {
        int k = kk * 4 + hs * 2;         // this lane-half's K pair within the chunk
        float a0 = 0.f, a1v = 0.f;
        if (full || node < N) {
            a0  = fmaxf(a1[(size_t)node * 16 + k]     + b1[k],     0.0f);
            a1v = fmaxf(a1[(size_t)node * 16 + k + 1] + b1[k + 1], 0.0f);
        }
        v2f A; A.x = a0; A.y = a1v;
        // B rows k, k+1; W2 is [16,2] row-major, cols >= 2 padded with 0
        float b0 = (lm < 2) ? W2[k * 2 + lm]       : 0.f;
        float bb = (lm < 2) ? W2[(k + 1) * 2 + lm] : 0.f;
        v2f B; B.x = b0; B.y = bb;
        C = __builtin_amdgcn_wmma_f32_16x16x4_f32(false, A, false, B, (short)0, C, false, false);
    }
    if (lm < 2) {
        #pragma unroll
        for (int r = 0; r < 8; ++r) {
            int row = m0 + r + hs * 8;
            if (full || row < N) {
                float v = C[r];
                float di = dinv[row];
                h2[(size_t)row * 2 + lm] = v;
                a2[(size_t)row * 2 + lm] = di * di * v;
            }
        }
    }
}

// ---------------- layer-2 edge scatter ----------------

__global__ void k_edges2(const int* __restrict__ src, const int* __restrict__ dst,
                         const float* __restrict__ dinv, const float* __restrict__ h,
                         float* __restrict__ acc, int E) {
    int e = blockIdx.x * blockDim.x + threadIdx.x;
    if (e >= E) return;
    int s = src[e], d = dst[e];
    float norm = dinv[s] * dinv[d];
    float2 v = *(const float2*)(h + (size_t)s * 2);
    atomicAdd(acc + (size_t)d * 2 + 0, v.x * norm);
    atomicAdd(acc + (size_t)d * 2 + 1, v.y * norm);
}

// ---------------- pooling + log_softmax ----------------

__global__ void k_zero_pool(float* __restrict__ sums, float* __restrict__ cnt) {
    int i = blockIdx.x * blockDim.x + threadIdx.x;
    if (i < NUM_GRAPHS * 2) sums[i] = 0.0f;
    if (i < NUM_GRAPHS) cnt[i] = 0.0f;
}

// batch is sorted: most waves see a single graph id. Wave-reduce then one atomic
// triplet per wave in the uniform case. No early returns so lane ops are defined.
__global__ void k_pool(const float* __restrict__ acc, const float* __restrict__ b2,
                       const int* __restrict__ batch, float* __restrict__ sums,
                       float* __restrict__ cnt, int N) {
    int i = blockIdx.x * blockDim.x + threadIdx.x;
    bool valid = (i < N);
    int g = valid ? batch[i] : -1;
    float v0 = valid ? acc[(size_t)i * 2 + 0] + b2[0] : 0.0f;
    float v1 = valid ? acc[(size_t)i * 2 + 1] + b2[1] : 0.0f;
    float c  = valid ? 1.0f : 0.0f;

    int g0 = __shfl(g, 0, 32);
    if (__all(g == g0)) {
        #pragma unroll
        for (int m = 16; m >= 1; m >>= 1) {
            v0 += __shfl_xor(v0, m, 32);
            v1 += __shfl_xor(v1, m, 32);
            c  += __shfl_xor(c,  m, 32);
        }
        if ((threadIdx.x & 31) == 0 && g0 >= 0) {
            atomicAdd(&sums[g0 * 2 + 0], v0);
            atomicAdd(&sums[g0 * 2 + 1], v1);
            atomicAdd(&cnt[g0], c);
        }
    } else if (valid) {
        atomicAdd(&sums[g * 2 + 0], v0);
        atomicAdd(&sums[g * 2 + 1], v1);
        atomicAdd(&cnt[g], 1.0f);
    }
}

__global__ void k_logsoftmax(const float* __restrict__ sums, const float* __restrict__ cnt,
                             float* __restrict__ out) {
    int g = blockIdx.x * blockDim.x + threadIdx.x;
    if (g >= NUM_GRAPHS) return;
    float c = fmaxf(cnt[g], 1.0f);
    float p0 = sums[g * 2 + 0] / c;
    float p1 = sums[g * 2 + 1] / c;
    float m = fmaxf(p0, p1);
    float lse = m + logf(expf(p0 - m) + expf(p1 - m));
    out[g * 2 + 0] = p0 - lse;
    out[g * 2 + 1] = p1 - lse;
}

// ---------------- launch ----------------

extern "C" void kernel_launch(void* const* d_in, const int* in_sizes, int n_in,
                              void* d_out, int out_size, void* d_ws, size_t ws_size,
                              hipStream_t stream) {
    const float* x  = (const float*)d_in[0];
    const int*   ei = (const int*)d_in[1];
    const int*   batch = (const int*)d_in[2];
    const float* W1 = (const float*)d_in[3];
    const float* b1 = (const float*)d_in[4];
    const float* W2 = (const float*)d_in[5];
    const float* b2 = (const float*)d_in[6];
    float* out = (float*)d_out;

    const int N = in_sizes[0] / 3;
    const int E = in_sizes[1] / 2;
    const int* src = ei;          // edge_index[0]
    const int* dst = ei + E;      // edge_index[1]

    // workspace layout (256B aligned)
    char* ws = (char*)d_ws;
    size_t off = 0;
    auto take = [&](size_t bytes) { size_t o = off; off = (off + bytes + 255) & ~(size_t)255; return o; };
    float* dinv = (float*)(ws + take((size_t)N * 4));            // u32 degree, then f32 dinv
    float* h1   = (float*)(ws + take((size_t)N * 16 * 4));       // x @ W1
    float* a1   = (float*)(ws + take((size_t)N * 16 * 4));       // self-loop init + edge atomics
    float* h2   = (float*)(ws + take((size_t)N * 2 * 4));        // relu(a1+b1) @ W2
    float* a2   = (float*)(ws + take((size_t)N * 2 * 4));        // self-loop init + edge atomics
    float* sums = (float*)(ws + take((size_t)NUM_GRAPHS * 2 * 4));
    float* cnt  = (float*)(ws + take((size_t)NUM_GRAPHS * 4));
    (void)ws_size; (void)n_in; (void)out_size;

    const int T = 256;
    const int gN = (N + T - 1) / T;
    const int gE = (E + T - 1) / T;
    // one wave (32 threads) per 16 node-rows -> 8 waves/block = 128 rows/block
    const int gW = ((N + 15) / 16 * 32 + T - 1) / T;

    // degree + norm
    k_deg_init <<<gN, T, 0, stream>>>((unsigned int*)dinv, N);
    k_deg_edges<<<gE, T, 0, stream>>>(dst, (unsigned int*)dinv, E);
    k_dinv     <<<gN, T, 0, stream>>>(dinv, N);

    // layer 1: fused GEMM + self-loop init, then edge scatter
    k_gemm1_wmma<<<gW, T, 0, stream>>>(x, W1, dinv, h1, a1, N);
    k_edges16   <<<gE, T, 0, stream>>>(src, dst, dinv, h1, a1, E);

    // layer 2: fused bias+ReLU+GEMM + self-loop init, then edge scatter
    k_gemm2_wmma<<<gW, T, 0, stream>>>(a1, b1, W2, dinv, h2, a2, N);
    k_edges2    <<<gE, T, 0, stream>>>(src, dst, dinv, h2, a2, E);

    // pooling + log_softmax
    k_zero_pool <<<(NUM_GRAPHS * 2 + T - 1) / T, T, 0, stream>>>(sums, cnt);
    k_pool      <<<gN, T, 0, stream>>>(a2, b2, batch, sums, cnt, N);
    k_logsoftmax<<<(NUM_GRAPHS + T - 1) / T, T, 0, stream>>>(sums, cnt, out);
}